// FcDomainGCN_62654982914338
// MI455X (gfx1250) — compile-verified
//
#include <hip/hip_runtime.h>
#include <hip/hip_bf16.h>

typedef __attribute__((ext_vector_type(2))) float v2f;
typedef __attribute__((ext_vector_type(8))) float v8f;

#define N_NODES  100000
#define N_EDGES  1600000
#define N_GRAPHS 256
#define HDIM     64
#define BN_EPS   1e-5f

// ---------------------------------------------------------------- utilities
__global__ void fill_zero_kernel(float* __restrict__ p, long long n) {
    long long i = (long long)blockIdx.x * blockDim.x + threadIdx.x;
    if (i < n) p[i] = 0.0f;
}

// deg[dst] += 1 for every edge
__global__ void deg_count_kernel(const int* __restrict__ dst,
                                 float* __restrict__ deg, int E) {
    int e = blockIdx.x * blockDim.x + threadIdx.x;
    if (e < E) atomicAdd(&deg[dst[e]], 1.0f);
}

// deg -> dinv = rsqrt(deg + 1)   (self-loop included)
__global__ void dinv_kernel(float* __restrict__ deg, int n) {
    int i = blockIdx.x * blockDim.x + threadIdx.x;
    if (i < n) deg[i] = rsqrtf(deg[i] + 1.0f);
}

// ---------------------------------------------------------------- WMMA GEMM K=64
// H[nrows,64] = X[nrows,64] @ W[64,64].  One wave -> 16 rows x 64 cols.
// W staged in LDS K-pair interleaved: Wl[((k>>1)*64 + c)*2 + (k&1)], so a B
// fragment {W[ka][c], W[ka+1][c]} is one aligned ds_load_b64 into an even
// VGPR pair (no repack movs before the WMMA).
__global__ void gemm64_wmma_kernel(const float* __restrict__ X,
                                   const float* __restrict__ W,
                                   float* __restrict__ H, int nrows) {
    __shared__ float Wl[HDIM * HDIM];
    for (int i = threadIdx.x; i < HDIM * HDIM; i += blockDim.x) {
        const int k = i >> 6, c = i & 63;
        Wl[(((k >> 1) * HDIM) + c) * 2 + (k & 1)] = W[i];
    }
    __syncthreads();

    const int lane = threadIdx.x & 31;
    const int wave = threadIdx.x >> 5;
    const int tile = blockIdx.x * (blockDim.x >> 5) + wave;
    const int row0 = tile * 16;
    if (row0 >= nrows) return;

    const int m     = lane & 15;
    const int khalf = (lane >> 4) << 1;   // 0 or 2
    const float2* Arow = (const float2*)(X + (long long)(row0 + m) * HDIM);
    const v2f*    Wp   = (const v2f*)Wl;  // Wp[p*64 + c] = {W[2p][c], W[2p+1][c]}

    v8f acc0 = {}, acc1 = {}, acc2 = {}, acc3 = {};

#pragma unroll
    for (int k0 = 0; k0 < HDIM; k0 += 4) {
        const int ka = k0 + khalf;             // even
        const float2 af = Arow[ka >> 1];       // X[row][ka], X[row][ka+1]
        v2f a;  a[0] = af.x;  a[1] = af.y;

        const v2f* Wrow = Wp + (ka >> 1) * HDIM;
        const v2f b0 = Wrow[m     ];
        const v2f b1 = Wrow[m + 16];
        const v2f b2 = Wrow[m + 32];
        const v2f b3 = Wrow[m + 48];

        acc0 = __builtin_amdgcn_wmma_f32_16x16x4_f32(false, a, false, b0, (short)0, acc0, false, false);
        acc1 = __builtin_amdgcn_wmma_f32_16x16x4_f32(false, a, false, b1, (short)0, acc1, false, false);
        acc2 = __builtin_amdgcn_wmma_f32_16x16x4_f32(false, a, false, b2, (short)0, acc2, false, false);
        acc3 = __builtin_amdgcn_wmma_f32_16x16x4_f32(false, a, false, b3, (short)0, acc3, false, false);
    }

    const int mbase = (lane < 16) ? 0 : 8;
#pragma unroll
    for (int v = 0; v < 8; ++v) {
        float* out = H + (long long)(row0 + mbase + v) * HDIM + m;
        out[ 0] = acc0[v];
        out[16] = acc1[v];
        out[32] = acc2[v];
        out[48] = acc3[v];
    }
}

// ---------------------------------------------------------------- WMMA GEMM K=3
// H[nrows,64] = X[nrows,3] @ W[3,64].  Same K-pair interleaved LDS layout,
// with the pad row k=3 zeroed during staging.
__global__ void gemm3_wmma_kernel(const float* __restrict__ X,
                                  const float* __restrict__ W,
                                  float* __restrict__ H, int nrows) {
    __shared__ float Wl[4 * HDIM];
    for (int i = threadIdx.x; i < 4 * HDIM; i += blockDim.x) {
        const int k = i >> 6, c = i & 63;
        Wl[(((k >> 1) * HDIM) + c) * 2 + (k & 1)] = (k < 3) ? W[i] : 0.0f;
    }
    __syncthreads();

    const int lane = threadIdx.x & 31;
    const int wave = threadIdx.x >> 5;
    const int tile = blockIdx.x * (blockDim.x >> 5) + wave;
    const int row0 = tile * 16;
    if (row0 >= nrows) return;

    const int m     = lane & 15;
    const int khalf = (lane >> 4) << 1;   // 0 or 2
    const int row   = row0 + m;

    // A: lanes 0-15 -> {X[r][0], X[r][1]}, lanes 16-31 -> {X[r][2], 0}
    const long long base = (long long)row * 3;
    const float a0 = X[base + khalf];                       // k=0 or k=2, valid
    long long i1 = base + khalf + 1;                        // k=1 or k=3(pad)
    if (i1 > (long long)nrows * 3 - 1) i1 = (long long)nrows * 3 - 1;
    const float a1v = X[i1];
    v2f a;  a[0] = a0;  a[1] = (khalf == 0) ? a1v : 0.0f;

    const v2f* Wrow = (const v2f*)Wl + (khalf >> 1) * HDIM;
    const v2f b0 = Wrow[m     ];
    const v2f b1 = Wrow[m + 16];
    const v2f b2 = Wrow[m + 32];
    const v2f b3 = Wrow[m + 48];

    v8f acc0 = {}, acc1 = {}, acc2 = {}, acc3 = {};
    acc0 = __builtin_amdgcn_wmma_f32_16x16x4_f32(false, a, false, b0, (short)0, acc0, false, false);
    acc1 = __builtin_amdgcn_wmma_f32_16x16x4_f32(false, a, false, b1, (short)0, acc1, false, false);
    acc2 = __builtin_amdgcn_wmma_f32_16x16x4_f32(false, a, false, b2, (short)0, acc2, false, false);
    acc3 = __builtin_amdgcn_wmma_f32_16x16x4_f32(false, a, false, b3, (short)0, acc3, false, false);

    const int mbase = (lane < 16) ? 0 : 8;
#pragma unroll
    for (int v = 0; v < 8; ++v) {
        float* out = H + (long long)(row0 + mbase + v) * HDIM + m;
        out[ 0] = acc0[v];
        out[16] = acc1[v];
        out[32] = acc2[v];
        out[48] = acc3[v];
    }
}

// ---------------------------------------------------------------- init agg
// AGG = H * dinv^2   (self-loop message; replaces the zero fill)
__global__ void init_agg_kernel(const float* __restrict__ H,
                                const float* __restrict__ dinv,
                                float* __restrict__ AGG) {
    long long gid = (long long)blockIdx.x * blockDim.x + threadIdx.x;
    if (gid >= (long long)N_NODES * HDIM) return;
    const float di = dinv[gid >> 6];
    AGG[gid] = H[gid] * di * di;
}

// ---------------------------------------------------------------- edge scatter
// AGG[dst] += H[src] * dinv[src]*dinv[dst]; one thread = 1 edge x 4 features
__global__ void edge_scatter_kernel(const float* __restrict__ H,
                                    const int* __restrict__ src,
                                    const int* __restrict__ dst,
                                    const float* __restrict__ dinv,
                                    float* __restrict__ AGG, int E) {
    long long gid = (long long)blockIdx.x * blockDim.x + threadIdx.x;
    if (gid >= (long long)E * 16) return;
    const int e = (int)(gid >> 4);
    const int f = ((int)gid & 15) << 2;
    const int s = src[e];
    const int d = dst[e];
    const float c = dinv[s] * dinv[d];
    const float4 h4 = *(const float4*)(H + (long long)s * HDIM + f);
    float* o = AGG + (long long)d * HDIM + f;
    atomicAdd(o + 0, h4.x * c);
    atomicAdd(o + 1, h4.y * c);
    atomicAdd(o + 2, h4.z * c);
    atomicAdd(o + 3, h4.w * c);
}

// ---------------------------------------------------------------- BN + ReLU
// OUT = relu(bn(AGG + b))   (self-loop already folded into AGG)
__global__ void bn_relu_kernel(const float* __restrict__ AGG,
                               const float* __restrict__ b,
                               const float* __restrict__ g,
                               const float* __restrict__ be,
                               const float* __restrict__ rm,
                               const float* __restrict__ rv,
                               float* __restrict__ OUT) {
    long long gid = (long long)blockIdx.x * blockDim.x + threadIdx.x;
    if (gid >= (long long)N_NODES * HDIM) return;
    const int c = (int)gid & 63;
    float v = AGG[gid] + b[c];
    v = (v - rm[c]) * rsqrtf(rv[c] + BN_EPS) * g[c] + be[c];
    OUT[gid] = fmaxf(v, 0.0f);
}

// ---------------------------------------------------------------- pooling
// x >= 0 after relu, so float bit pattern is monotone under u32 atomicMax.
__global__ void pool_kernel(const float* __restrict__ X,
                            const int* __restrict__ bidx,
                            float* __restrict__ psum,
                            unsigned* __restrict__ pmax,
                            float* __restrict__ pcnt) {
    long long gid = (long long)blockIdx.x * blockDim.x + threadIdx.x;
    if (gid >= (long long)N_NODES * HDIM) return;
    const int n = (int)(gid >> 6);
    const int c = (int)gid & 63;
    const int b = bidx[n];
    const float v = X[gid];
    atomicAdd(&psum[b * HDIM + c], v);
    atomicMax(&pmax[b * HDIM + c], __float_as_uint(v));
    if (c == 0) atomicAdd(&pcnt[b], 1.0f);
}

// ---------------------------------------------------------------- MLP head
// One block (128 threads) per graph; xg = [mean(64) | max(64)]
__global__ void mlp_kernel(const float* __restrict__ psum,
                           const unsigned* __restrict__ pmax,
                           const float* __restrict__ pcnt,
                           const float* __restrict__ mW1, const float* __restrict__ mb1,
                           const float* __restrict__ mW2, const float* __restrict__ mb2,
                           const float* __restrict__ mW3, const float* __restrict__ mb3,
                           float* __restrict__ out) {
    __shared__ float xg[2 * HDIM];
    __shared__ float h1[32];
    __shared__ float h2[16];
    const int gph = blockIdx.x;
    const int t   = threadIdx.x;

    const float cnt = fmaxf(pcnt[gph], 1.0f);
    if (t < HDIM)          xg[t] = psum[gph * HDIM + t] / cnt;
    else if (t < 2 * HDIM) xg[t] = __uint_as_float(pmax[gph * HDIM + (t - HDIM)]);
    __syncthreads();

    if (t < 32) {
        float a = mb1[t];
#pragma unroll 8
        for (int i = 0; i < 2 * HDIM; ++i) a += xg[i] * mW1[i * 32 + t];
        h1[t] = fmaxf(a, 0.0f);
    }
    __syncthreads();
    if (t < 16) {
        float a = mb2[t];
#pragma unroll
        for (int i = 0; i < 32; ++i) a += h1[i] * mW2[i * 16 + t];
        h2[t] = fmaxf(a, 0.0f);
    }
    __syncthreads();
    if (t == 0) {
        float a = mb3[0];
#pragma unroll
        for (int i = 0; i < 16; ++i) a += h2[i] * mW3[i];
        out[gph] = a;
    }
}

// ---------------------------------------------------------------- launcher
extern "C" void kernel_launch(void* const* d_in, const int* in_sizes, int n_in,
                              void* d_out, int out_size, void* d_ws, size_t ws_size,
                              hipStream_t stream) {
    const float* x    = (const float*)d_in[0];
    const int*   ei   = (const int*)d_in[1];
    const int*   bidx = (const int*)d_in[2];
    const float* W[3]  = {(const float*)d_in[3],  (const float*)d_in[9],  (const float*)d_in[15]};
    const float* B[3]  = {(const float*)d_in[4],  (const float*)d_in[10], (const float*)d_in[16]};
    const float* Gm[3] = {(const float*)d_in[5],  (const float*)d_in[11], (const float*)d_in[17]};
    const float* Be[3] = {(const float*)d_in[6],  (const float*)d_in[12], (const float*)d_in[18]};
    const float* Rm[3] = {(const float*)d_in[7],  (const float*)d_in[13], (const float*)d_in[19]};
    const float* Rv[3] = {(const float*)d_in[8],  (const float*)d_in[14], (const float*)d_in[20]};
    const float* mW1 = (const float*)d_in[21];
    const float* mb1 = (const float*)d_in[22];
    const float* mW2 = (const float*)d_in[23];
    const float* mb2 = (const float*)d_in[24];
    const float* mW3 = (const float*)d_in[25];
    const float* mb3 = (const float*)d_in[26];

    const int* src = ei;
    const int* dst = ei + N_EDGES;

    // workspace layout (all 16B aligned: N_NODES*4 = 400000 bytes % 16 == 0)
    float*    ws   = (float*)d_ws;
    float*    dinv = ws;                               // N
    float*    bufH = dinv + N_NODES;                   // N*64
    float*    bufX = bufH + (long long)N_NODES * HDIM; // N*64
    float*    bufA = bufX + (long long)N_NODES * HDIM; // N*64
    float*    psum = bufA + (long long)N_NODES * HDIM; // G*64
    unsigned* pmax = (unsigned*)(psum + N_GRAPHS * HDIM); // G*64
    float*    pcnt = (float*)(pmax + N_GRAPHS * HDIM);    // G

    const long long NF = (long long)N_NODES * HDIM;
    const int TB = 256;
    auto blocks = [](long long n, int tb) { return (int)((n + tb - 1) / tb); };

    // degrees -> dinv
    fill_zero_kernel<<<blocks(N_NODES, TB), TB, 0, stream>>>(dinv, N_NODES);
    deg_count_kernel<<<blocks(N_EDGES, TB), TB, 0, stream>>>(dst, dinv, N_EDGES);
    dinv_kernel<<<blocks(N_NODES, TB), TB, 0, stream>>>(dinv, N_NODES);

    const int rowTiles    = N_NODES / 16;            // 6250 (exact)
    const int wavesPerBlk = TB / 32;                 // 8
    const int gemmBlocks  = (rowTiles + wavesPerBlk - 1) / wavesPerBlk;

    const float* curX = x;
    for (int l = 0; l < 3; ++l) {
        if (l == 0)
            gemm3_wmma_kernel<<<gemmBlocks, TB, 0, stream>>>(curX, W[l], bufH, N_NODES);
        else
            gemm64_wmma_kernel<<<gemmBlocks, TB, 0, stream>>>(curX, W[l], bufH, N_NODES);
        init_agg_kernel<<<blocks(NF, TB), TB, 0, stream>>>(bufH, dinv, bufA);
        edge_scatter_kernel<<<blocks((long long)N_EDGES * 16, TB), TB, 0, stream>>>(
            bufH, src, dst, dinv, bufA, N_EDGES);
        bn_relu_kernel<<<blocks(NF, TB), TB, 0, stream>>>(
            bufA, B[l], Gm[l], Be[l], Rm[l], Rv[l], bufX);
        curX = bufX;
    }

    // pooling buffers (pmax zero-bits == 0.0f, valid since activations >= 0)
    fill_zero_kernel<<<blocks(N_GRAPHS * HDIM, TB), TB, 0, stream>>>(psum, N_GRAPHS * HDIM);
    fill_zero_kernel<<<blocks(N_GRAPHS * HDIM, TB), TB, 0, stream>>>((float*)pmax, N_GRAPHS * HDIM);
    fill_zero_kernel<<<blocks(N_GRAPHS, TB), TB, 0, stream>>>(pcnt, N_GRAPHS);
    pool_kernel<<<blocks(NF, TB), TB, 0, stream>>>(bufX, bidx, psum, pmax, pcnt);

    mlp_kernel<<<N_GRAPHS, 128, 0, stream>>>(psum, pmax, pcnt,
                                             mW1, mb1, mW2, mb2, mW3, mb3,
                                             (float*)d_out);
}